// Adapthisteq_39436389711988
// MI455X (gfx1250) — compile-verified
//
#include <hip/hip_runtime.h>

// Reference analysis: TILE=6 -> npix=36 -> step = (36 - cnt_last)//255 == 0 for
// every tile -> jnp.where(step==0, ...) always selects the ORIGINAL tiles.
// The whole pipeline is the identity on `pic`; the only real work is the
// int32 -> float32 dtype conversion for the harness output buffer.
//
// MI455X plan: streaming copy-convert at HBM rate.
//   traffic = 50.2 MB read + 50.2 MB write -> ~4.3 us cold at 23.3 TB/s.
//   Loads: default RT hint -> input stays resident in the 192 MB L2 across
//   graph replays (50 MB << 192 MB). Stores: non-temporal -> write stream
//   doesn't evict it. Steady-state replay ~ write-bandwidth bound (~2.2 us).
//   No matrix math exists (identity transform), so WMMA is inapplicable;
//   the optimal path is the 128-bit VMEM pipe.
//
// The vector kernel handles only full vec4 groups (single guard, load issues
// immediately); any ragged tail (never present for the reference shape,
// n % 4 == 0) is handled by a separate scalar kernel so no cold tail code
// pollutes the hot kernel.

typedef __attribute__((ext_vector_type(4))) int   i32x4;
typedef __attribute__((ext_vector_type(4))) float f32x4;

__global__ __launch_bounds__(256) void adapthisteq_copy_vec4(
    const int* __restrict__ in, float* __restrict__ out, int n4)
{
    int vid = blockIdx.x * 256 + threadIdx.x;   // vec4 group index, 32-bit
    if (vid < n4) {
        int base = vid * 4;
        // global_load_b128, RT temporal hint (L2-resident across replays)
        i32x4 v = *reinterpret_cast<const i32x4*>(in + base);
        f32x4 f;
        f.x = (float)v.x;   // v_cvt_f32_i32 x4
        f.y = (float)v.y;
        f.z = (float)v.z;
        f.w = (float)v.w;
        // global_store_b128 TH=NT: streaming write, don't pollute L2
        __builtin_nontemporal_store(f, reinterpret_cast<f32x4*>(out + base));
    }
}

__global__ __launch_bounds__(32) void adapthisteq_copy_tail(
    const int* __restrict__ in, float* __restrict__ out, int start, int n)
{
    int i = start + (int)threadIdx.x;
    if (i < n) {
        out[i] = (float)in[i];
    }
}

extern "C" void kernel_launch(void* const* d_in, const int* in_sizes, int n_in,
                              void* d_out, int out_size, void* d_ws, size_t ws_size,
                              hipStream_t stream)
{
    (void)n_in; (void)d_ws; (void)ws_size; (void)out_size;

    const int* in  = (const int*)d_in[0];
    float*     out = (float*)d_out;
    const int  n   = in_sizes[0];      // 2046*2046*3 = 12,558,348 (n % 4 == 0)

    const int n4    = n / 4;           // full vec4 groups
    const int block = 256;             // 8 wave32s per block
    const int grid  = (n4 + block - 1) / block;

    if (grid > 0) {
        adapthisteq_copy_vec4<<<grid, block, 0, stream>>>(in, out, n4);
    }
    const int tail = n - n4 * 4;       // 0 for the reference shape
    if (tail > 0) {
        adapthisteq_copy_tail<<<1, 32, 0, stream>>>(in, out, n4 * 4, n);
    }
}